// AdjacencySmoothnessLoss_87660282511505
// MI455X (gfx1250) — compile-verified
//
#include <hip/hip_runtime.h>
#include <hip/hip_bf16.h>

#define N            8192
#define INTER_MAX    (N - 2)          // last valid interior index = 8190
#define BLOCK_THREADS 256
#define ROWS_PER_BLOCK 16
#define COL_TILES    (N / 4)          // 2048 four-column slabs
#define GRID_X       (COL_TILES / BLOCK_THREADS)            // 8
#define GRID_Y       ((N - 2 + ROWS_PER_BLOCK - 1) / ROWS_PER_BLOCK) // 512
#define NBLOCKS      (GRID_X * GRID_Y)                      // 4096
#define LAMBDA_SMOOTH 0.1f

typedef __attribute__((ext_vector_type(2))) float v2f;
typedef __attribute__((ext_vector_type(8))) float v8f;

// ds_swizzle group-of-32 encodings: offset[14:10]=xor, [9:5]=or, [4:0]=and
#define SWZ_SWAPX1 0x041F
#define SWZ_SWAPX2 0x081F
#define SWZ_SWAPX4 0x101F
#define SWZ_SWAPX8 0x201F

// Reduce one f32 value per lane (wave32) to a wave total using
// V_WMMA_F32_16X16X4_F32 with an all-ones A matrix:
//   D[m][n] = sum_k 1 * B[k][n]  -> every row of D holds the 16 column sums.
// The 32 per-lane partials occupy 32 distinct B slots (b[0]); b[1] = 0.
// After the WMMA, c[0] in lanes 0..15 holds colsum[n] (lanes 16..31 hold the
// duplicate M=8 row); 4 xor-swizzle steps within each 16-lane half finish the
// reduction. EXEC must be all 1s at the call site.
__device__ __forceinline__ float wave_reduce_wmma(float p) {
    v2f a; a[0] = 1.0f; a[1] = 1.0f;   // A(16x4) = all ones
    v2f b; b[0] = p;    b[1] = 0.0f;   // per-lane partial in a unique B slot
    v8f c = {};
    c = __builtin_amdgcn_wmma_f32_16x16x4_f32(
            /*neg_a=*/false, a, /*neg_b=*/false, b,
            /*c_mod=*/(short)0, c, /*reuse_a=*/false, /*reuse_b=*/false);
    float t = c[0];
    t += __int_as_float(__builtin_amdgcn_ds_swizzle(__float_as_int(t), SWZ_SWAPX1));
    t += __int_as_float(__builtin_amdgcn_ds_swizzle(__float_as_int(t), SWZ_SWAPX2));
    t += __int_as_float(__builtin_amdgcn_ds_swizzle(__float_as_int(t), SWZ_SWAPX4));
    t += __int_as_float(__builtin_amdgcn_ds_swizzle(__float_as_int(t), SWZ_SWAPX8));
    return t;  // lane 0 (and lane 16) hold the wave total
}

__global__ __launch_bounds__(BLOCK_THREADS)
void smooth_partial_kernel(const float* __restrict__ adj, float* __restrict__ part) {
    const int j  = blockIdx.x * BLOCK_THREADS + threadIdx.x;  // 4-col slab index
    const int c0 = j * 4;                                     // 16B aligned base

    // Branch-free clamped edge offsets (uniform per thread, loop-invariant).
    // When the clamp kicks in, the loaded value feeds only an element whose
    // contribution is masked invalid below, so the value doesn't matter --
    // only address safety does.
    const int loff = (c0 > 0)     ? -1 : 0;
    const int roff = (c0 + 4 < N) ?  4 : 0;

    const int r0   = 1 + blockIdx.y * ROWS_PER_BLOCK;
    const int rEnd = min(r0 + ROWS_PER_BLOCK, INTER_MAX + 1); // uniform trip count

    float p = 0.0f;

    #pragma unroll 4
    for (int r = r0; r < rEnd; ++r) {
        const size_t rowU = (size_t)(r - 1) * N + c0;
        const size_t rowC = (size_t)(r    ) * N + c0;
        const size_t rowD = (size_t)(r + 1) * N + c0;
        const float4 up = *(const float4*)(adj + rowU);
        const float4 ce = *(const float4*)(adj + rowC);
        const float4 dn = *(const float4*)(adj + rowD);
        const float lft = adj[rowC + loff];   // unconditional, clamped address
        const float rgt = adj[rowC + roff];   // unconditional, clamped address

        const float cv[4] = {ce.x, ce.y, ce.z, ce.w};
        const float uv[4] = {up.x, up.y, up.z, up.w};
        const float dv[4] = {dn.x, dn.y, dn.z, dn.w};

        #pragma unroll
        for (int i = 0; i < 4; ++i) {
            const int   c  = c0 + i;
            const float l  = (i == 0) ? lft : cv[i - 1];
            const float rg = (i == 3) ? rgt : cv[i + 1];
            const float navg = (uv[i] + dv[i] + l + rg) * 0.25f;
            const float val  = fabsf(cv[i] - navg);
            // value-zeroing instead of divergence: EXEC stays all-ones
            p += (c >= 1 && c <= INTER_MAX) ? val : 0.0f;
        }
    }

    // all lanes converged here; EXEC is full for the WMMA reduction
    const float wsum = wave_reduce_wmma(p);

    __shared__ float s[BLOCK_THREADS / 32];
    const int lane = threadIdx.x & 31;
    const int wv   = threadIdx.x >> 5;
    if (lane == 0) s[wv] = wsum;
    __syncthreads();
    if (threadIdx.x == 0) {
        float t = 0.0f;
        #pragma unroll
        for (int i = 0; i < BLOCK_THREADS / 32; ++i) t += s[i];
        part[blockIdx.y * GRID_X + blockIdx.x] = t;
    }
}

__global__ __launch_bounds__(BLOCK_THREADS)
void smooth_final_kernel(const float* __restrict__ part, float* __restrict__ out, int n) {
    float p = 0.0f;
    for (int i = threadIdx.x; i < n; i += BLOCK_THREADS) p += part[i];

    const float wsum = wave_reduce_wmma(p);

    __shared__ float s[BLOCK_THREADS / 32];
    const int lane = threadIdx.x & 31;
    const int wv   = threadIdx.x >> 5;
    if (lane == 0) s[wv] = wsum;
    __syncthreads();
    if (threadIdx.x == 0) {
        float t = 0.0f;
        #pragma unroll
        for (int i = 0; i < BLOCK_THREADS / 32; ++i) t += s[i];
        out[0] = LAMBDA_SMOOTH * t;
    }
}

extern "C" void kernel_launch(void* const* d_in, const int* in_sizes, int n_in,
                              void* d_out, int out_size, void* d_ws, size_t ws_size,
                              hipStream_t stream) {
    const float* adj  = (const float*)d_in[0];
    float*       part = (float*)d_ws;          // NBLOCKS floats = 16 KiB scratch
    float*       out  = (float*)d_out;

    dim3 grid(GRID_X, GRID_Y);
    smooth_partial_kernel<<<grid, BLOCK_THREADS, 0, stream>>>(adj, part);
    smooth_final_kernel<<<1, BLOCK_THREADS, 0, stream>>>(part, out, NBLOCKS);
}